// CTLSTM_2929167696632
// MI455X (gfx1250) — compile-verified
//
#include <hip/hip_runtime.h>

typedef __attribute__((ext_vector_type(2))) float v2f;
typedef __attribute__((ext_vector_type(8))) float v8f;

#define BATCH  256
#define MAXLEN 512
#define EMBED  128
#define HID    256
#define FANIN  384    // EMBED + HID
#define NOUT   1792   // 7 * HID
#define BTILE  16
#define AROW   388    // FANIN + 4 pad: 16B-aligned rows, conflict-free b64 A-frag reads

#define NSLAB   (FANIN / 4)     // 96  K-slabs of 4
#define NPAIR   (NSLAB / 2)     // 48  K-slab pairs (b128 packed granularity)
#define NTILE   (NOUT / 16)     // 112 N-tiles
#define PACKELTS ((size_t)FANIN * NOUT)          // 688128 floats
#define PACKBYTES (PACKELTS * sizeof(float))     // 2,752,512 bytes

__device__ __forceinline__ float sigf(float z) {
    return 1.0f / (1.0f + __expf(-z));
}

// ---------------------------------------------------------------------------
// One-time repack of W (384x1792 row-major) into WMMA B-fragment order:
//   Wp[((p*112 + j)*32 + lane)*4 + e], e=0,1 -> K-slab 2p, e=2,3 -> K-slab 2p+1
//   lane<16 covers K rows {4s, 4s+1}, lane>=16 covers {4s+2, 4s+3}, col = 16j+lane%16
// so the scan kernel fetches one b128 per lane per (slab-pair, n-tile).
// ---------------------------------------------------------------------------
__global__ void pack_w_kernel(const float* __restrict__ W, float* __restrict__ Wp) {
    int i = blockIdx.x * blockDim.x + threadIdx.x;   // exact launch: PACKELTS threads
    int e    = i & 3;
    int lane = (i >> 2) & 31;
    int jt   = (i >> 7) % NTILE;
    int p    = (i >> 7) / NTILE;
    int slab = 2 * p + (e >> 1);
    int r    = 2 * (lane >> 4) + (e & 1);
    int k    = 4 * slab + r;
    int n    = 16 * jt + (lane & 15);
    Wp[i] = W[(size_t)k * NOUT + n];
}

template <bool PACKED>
__launch_bounds__(512, 1)
__global__ void ctlstm_scan_kernel(const float* __restrict__ embeds,   // (B, T, E)
                                   const float* __restrict__ dtimes,   // (B, T)
                                   const float* __restrict__ W,        // (384, 1792)
                                   const float* __restrict__ Wp,       // packed (or null)
                                   const float* __restrict__ bias,     // (1792,)
                                   float* __restrict__ out)            // 5 x (B, T, H)
{
    __shared__ float Abuf[BTILE][AROW];        // [x_t (128) | dh (256)] per batch row
    __shared__ float dtall[BTILE * MAXLEN];    // this block's dtimes rows

    const int tid  = threadIdx.x;
    const int lane = tid & 31;
    const int wv   = tid >> 5;        // wave 0..15 -> hidden slice [16*wv, 16*wv+16)
    const int half = lane >> 4;       // 0: lanes 0-15, 1: lanes 16-31
    const int nl   = lane & 15;
    const int b0   = blockIdx.x * BTILE;
    const int hcol = wv * 16 + nl;    // hidden index owned by this lane (all 8 D-rows)

    // ---- preload dtimes rows for this batch tile (coalesced float4) ----
    #pragma unroll
    for (int j = 0; j < 4; ++j) {
        int flat = tid * 4 + j * 2048;                    // 0..8191
        int r = flat >> 9, c = flat & 511;
        *(float4*)&dtall[flat] =
            *(const float4*)&dtimes[(size_t)(b0 + r) * MAXLEN + c];
    }
    // ---- zero the recurrent-h region of A (dh(t=-1) = 0) ----
    #pragma unroll
    for (int j = 0; j < 8; ++j) {
        int flat = tid + 512 * j;                         // 0..4095
        Abuf[flat >> 8][EMBED + (flat & 255)] = 0.0f;
    }

    // ---- per-lane persistent recurrent state: dc, cbar for 8 (m, hcol) pairs ----
    float dcS[8], cbS[8];
    #pragma unroll
    for (int v = 0; v < 8; ++v) { dcS[v] = 0.0f; cbS[v] = 0.0f; }

    // ---- bias for this lane's 7 gate-chunk columns ----
    float bz[7];
    #pragma unroll
    for (int c = 0; c < 7; ++c) bz[c] = bias[c * HID + hcol];

    const size_t S = (size_t)BATCH * MAXLEN * HID;
    float* outc  = out;          // cells
    float* outcb = out + S;      // cell_bars
    float* outdl = out + 2 * S;  // deltas
    float* outog = out + 3 * S;  // output_gates
    float* outdh = out + 4 * S;  // decayed_hs

    // x staging map: each thread moves 16B (4 floats) of the 16x128 tile
    const int xm = tid >> 5;            // batch row 0..15
    const int xc = (tid & 31) * 4;      // embed col 0..124
    const unsigned xlds = (unsigned)(uintptr_t)(void*)&Abuf[xm][xc];  // LDS byte offset

    const int krow0 = 2 * half;         // lane's K sub-rows within each K=4 slab

    // packed-W per-lane base: fragment stream for this wave's 7 chunk tiles
    const float* wpb = PACKED ? (Wp + ((size_t)wv * 32 + lane) * 4) : nullptr;

    for (int t = 0; t < MAXLEN; ++t) {
        // ---- async-stage x_t into A[:, 0:128] (global -> LDS, no VGPR bounce) ----
        const float* xsrc = embeds + ((size_t)(b0 + xm) * MAXLEN + t) * EMBED + xc;
        asm volatile("global_load_async_to_lds_b128 %0, %1, off"
                     :: "v"(xlds), "v"(xsrc) : "memory");
        if (t + 1 < MAXLEN)
            __builtin_prefetch(xsrc + EMBED, 0, 1);       // next step's x line
        asm volatile("s_wait_asynccnt 0x0" ::: "memory"); // x_t resident in LDS
        __syncthreads();   // x_t staged; dh(t-1) writes visible to all waves

        // ---- z = A(16x384) @ W(384x1792), f32 WMMA path; this wave's 7 N-tiles ----
        v8f acc[7] = {};
        if (PACKED) {
            for (int p = 0; p < NPAIR; ++p) {
                v2f a0 = *(const v2f*)&Abuf[nl][8 * p + krow0];       // slab 2p
                v2f a1 = *(const v2f*)&Abuf[nl][8 * p + 4 + krow0];   // slab 2p+1
                const float* wp = wpb + (size_t)p * (NTILE * 128);    // 112 tiles * 32 lanes * 4
                #pragma unroll
                for (int c = 0; c < 7; ++c) {
                    const float4 w4 = *(const float4*)(wp + c * (16 * 128)); // b128
                    v2f bA = {w4.x, w4.y};
                    v2f bB = {w4.z, w4.w};
                    acc[c] = __builtin_amdgcn_wmma_f32_16x16x4_f32(
                        false, a0, false, bA, (short)0, acc[c], false, false);
                    acc[c] = __builtin_amdgcn_wmma_f32_16x16x4_f32(
                        false, a1, false, bB, (short)0, acc[c], false, false);
                }
            }
        } else {
            for (int k0 = 0; k0 < FANIN; k0 += 4) {
                v2f afrag = *(const v2f*)&Abuf[nl][k0 + krow0];       // ds_load_b64
                const float* wrow = W + (size_t)(k0 + krow0) * NOUT + hcol;
                #pragma unroll
                for (int c = 0; c < 7; ++c) {
                    v2f bfrag;
                    bfrag.x = wrow[c * HID];
                    bfrag.y = wrow[c * HID + NOUT];
                    acc[c] = __builtin_amdgcn_wmma_f32_16x16x4_f32(
                        false, afrag, false, bfrag, (short)0, acc[c], false, false);
                }
            }
        }

        __syncthreads();   // every wave finished reading A before dh(t) overwrites it

        // ---- register-resident gates + state update + outputs ----
        #pragma unroll
        for (int v = 0; v < 8; ++v) {
            const int m = v + 8 * half;                    // D-tile row -> batch row
            const float dt  = dtall[m * MAXLEN + t];

            const float ig  = sigf(acc[0][v] + bz[0]);
            const float igb = sigf(acc[1][v] + bz[1]);
            const float fg  = sigf(acc[2][v] + bz[2]);
            const float fgb = sigf(acc[3][v] + bz[3]);
            const float og  = sigf(acc[4][v] + bz[4]);
            const float ti  = 2.0f * sigf(acc[5][v] + bz[5]) - 1.0f;

            const float zd    = acc[6][v] + bz[6];
            const float delta = fmaxf(zd, 0.0f) + __logf(1.0f + __expf(-fabsf(zd)));

            const float cell = fg  * dcS[v] + ig  * ti;
            const float cbar = fgb * cbS[v] + igb * ti;
            const float dcn  = cbar + (cell - cbar) * __expf(-delta * dt);
            const float dhn  = og * (2.0f * sigf(2.0f * dcn) - 1.0f);  // og * tanh(dcn)

            dcS[v] = dcn;
            cbS[v] = cbar;

            const size_t o = ((size_t)(b0 + m) * MAXLEN + t) * HID + hcol;
            outc[o]  = cell;
            outcb[o] = cbar;
            outdl[o] = delta;
            outog[o] = og;
            outdh[o] = dhn;

            Abuf[m][EMBED + hcol] = dhn;   // dh(t) for next step's GEMM
        }
        // next iteration's post-stage __syncthreads orders dh writes vs GEMM reads
    }
}

extern "C" void kernel_launch(void* const* d_in, const int* in_sizes, int n_in,
                              void* d_out, int out_size, void* d_ws, size_t ws_size,
                              hipStream_t stream) {
    const float* embeds = (const float*)d_in[0];
    const float* dtimes = (const float*)d_in[1];
    const float* W      = (const float*)d_in[2];
    const float* b      = (const float*)d_in[3];
    float* out          = (float*)d_out;
    (void)in_sizes; (void)n_in; (void)out_size;

    if (ws_size >= PACKBYTES) {
        float* Wp = (float*)d_ws;
        pack_w_kernel<<<dim3((unsigned)(PACKELTS / 256)), dim3(256), 0, stream>>>(W, Wp);
        ctlstm_scan_kernel<true><<<dim3(BATCH / BTILE), dim3(512), 0, stream>>>(
            embeds, dtimes, W, Wp, b, out);
    } else {
        ctlstm_scan_kernel<false><<<dim3(BATCH / BTILE), dim3(512), 0, stream>>>(
            embeds, dtimes, W, nullptr, b, out);
    }
}